// CenterLoss_27255862460722
// MI455X (gfx1250) — compile-verified
//
#include <hip/hip_runtime.h>
#include <stdint.h>

// CenterLoss: d_i = ||x_i||^2 + ||c_{y_i}||^2 - 2 x_i . c_{y_i}, clamped,
// mean over rows + (C_OUT-1)*1e-12.  Memory-bound gather+reduce.
//
// gfx1250 paths used:
//  - global_load_async_to_lds_b128 (ASYNCcnt) for the sequential x stream
//  - s_wait_asynccnt 0 before consuming the staged row
//  - global_prefetch_b8 for the next gathered center row
//  - wave32 butterfly reduction, split wait counters

#define DIM       128
#define ROWS      32768
#define NUM_C     100000
#define TPB       256            // 8 wave32 waves per block
#define WPB       (TPB / 32)
#define NBLOCKS   2048
#define CLAMP_LO  1e-12f
#define CLAMP_HI  1e12f

__global__ __launch_bounds__(TPB)
void center_loss_partial(const float* __restrict__ x,
                         const int*  __restrict__ labels,
                         const float* __restrict__ centers,
                         float* __restrict__ partial)
{
    // One 512B x-row staging slot per wave (async copy target) + block sums.
    __shared__ float4 stage[WPB * 32];
    __shared__ float  smem[WPB];

    const int lane   = threadIdx.x & 31;
    const int wave   = threadIdx.x >> 5;
    const int gwave  = blockIdx.x * WPB + wave;
    const int nwaves = gridDim.x * WPB;

    // Per-lane fixed LDS destination: low 32 bits of the flat shared-space
    // address are the workgroup-relative LDS byte offset on gfx1250.
    float4* const slot = &stage[wave * 32 + lane];
    const unsigned lds_off = (unsigned)(uintptr_t)(void*)slot;

    float acc = 0.0f;

    for (int r = gwave; r < ROWS; r += nwaves) {
        // ---- async: start copying this row of x into our LDS slot --------
        // Each lane moves 16B: x[r*128 + lane*4 .. +3] -> stage slot.
        const unsigned xoff = (unsigned)r * (DIM * 4) + (unsigned)lane * 16;
        asm volatile("global_load_async_to_lds_b128 %0, %1, %2"
                     :: "v"(lds_off), "v"(xoff), "s"(x)
                     : "memory");

        // ---- overlapped work: label fetch, prefetch, center gather -------
        const int lbl = __builtin_amdgcn_readfirstlane(labels[r]);

        const int rn = r + nwaves;
        if (rn < ROWS) {
            const int lbln = __builtin_amdgcn_readfirstlane(labels[rn]);
            __builtin_prefetch(centers + (size_t)lbln * DIM + lane * 4, 0, 3);
        }

        const float4 cv = *(const float4*)(centers + (size_t)lbl * DIM + lane * 4);

        float scc = cv.x * cv.x;
        scc = fmaf(cv.y, cv.y, scc);
        scc = fmaf(cv.z, cv.z, scc);
        scc = fmaf(cv.w, cv.w, scc);

        // ---- consume staged x row ----------------------------------------
        asm volatile("s_wait_asynccnt 0x0" ::: "memory");
        const float4 xv = *slot;   // ds_load_b128 from our own slot

        float sxx = xv.x * xv.x;
        sxx = fmaf(xv.y, xv.y, sxx);
        sxx = fmaf(xv.z, xv.z, sxx);
        sxx = fmaf(xv.w, xv.w, sxx);

        float sxc = xv.x * cv.x;
        sxc = fmaf(xv.y, cv.y, sxc);
        sxc = fmaf(xv.z, cv.z, sxc);
        sxc = fmaf(xv.w, cv.w, sxc);

        // Clamp applies to the full row sum; combination is linear, so
        // reduce a single per-lane value.
        float s = sxx + scc - 2.0f * sxc;

        // wave32 butterfly reduction
        #pragma unroll
        for (int m = 16; m >= 1; m >>= 1)
            s += __shfl_xor(s, m, 32);

        s = fminf(fmaxf(s, CLAMP_LO), CLAMP_HI);
        acc += s;                 // identical in every lane
    }

    if (lane == 0) smem[wave] = acc;
    __syncthreads();

    if (threadIdx.x == 0) {
        float b = 0.0f;
        #pragma unroll
        for (int w = 0; w < WPB; ++w) b += smem[w];
        partial[blockIdx.x] = b;  // fixed order -> deterministic
    }
}

__global__ __launch_bounds__(1024)
void center_loss_final(const float* __restrict__ partial, int n,
                       float* __restrict__ out)
{
    __shared__ float smem[32];

    float s = 0.0f;
    for (int i = threadIdx.x; i < n; i += 1024)
        s += partial[i];

    #pragma unroll
    for (int m = 16; m >= 1; m >>= 1)
        s += __shfl_xor(s, m, 32);

    const int lane = threadIdx.x & 31;
    const int wave = threadIdx.x >> 5;
    if (lane == 0) smem[wave] = s;
    __syncthreads();

    if (threadIdx.x == 0) {
        float t = 0.0f;
        #pragma unroll
        for (int w = 0; w < 32; ++w) t += smem[w];
        const float kconst = (float)((NUM_C - 1) * 1e-12);   // 9.9999e-8
        out[0] = t * (1.0f / (float)ROWS) + kconst;
    }
}

extern "C" void kernel_launch(void* const* d_in, const int* in_sizes, int n_in,
                              void* d_out, int out_size, void* d_ws, size_t ws_size,
                              hipStream_t stream)
{
    const float* x       = (const float*)d_in[0];
    const int*   labels  = (const int*)  d_in[1];
    const float* centers = (const float*)d_in[2];
    float*       out     = (float*)d_out;
    float*       partial = (float*)d_ws;

    // Clamp grid if the workspace is unexpectedly small (4 bytes per block).
    int nblocks = NBLOCKS;
    if (ws_size < (size_t)nblocks * sizeof(float))
        nblocks = (int)(ws_size / sizeof(float));
    if (nblocks < 1) nblocks = 1;

    center_loss_partial<<<nblocks, TPB, 0, stream>>>(x, labels, centers, partial);
    center_loss_final<<<1, 1024, 0, stream>>>(partial, nblocks, out);
}